// DEC_73882027426138
// MI455X (gfx1250) — compile-verified
//
#include <hip/hip_runtime.h>

typedef float v2f __attribute__((ext_vector_type(2)));
typedef float v8f __attribute__((ext_vector_type(8)));

#define N_ROWS 400000
#define DIMS 64
#define KC 128
#define ROW_STRIDE 68   // padded LDS row stride in dwords (68*4=272=17*16 -> 16B-aligned rows, conflict-free frag reads)

// ---------------------------------------------------------------------------
// Pass 1: per 128-row block: q = rownorm(1/(1+max(sq,0))), sq = |e|^2+|c|^2-2 e.c
// GEMM via V_WMMA_F32_16X16X4_F32. Also emits per-block column partial sums.
// ---------------------------------------------------------------------------
__global__ __launch_bounds__(256) void dec_pass1(const float* __restrict__ embeds,
                                                 const float* __restrict__ centers,
                                                 float* __restrict__ qout,      // d_out + 1
                                                 float* __restrict__ colpart)   // [gridDim.x * 128]
{
    __shared__ float lds_e[128 * ROW_STRIDE];  // 34816 B
    __shared__ float e2[128];
    __shared__ float c2[128];
    __shared__ float wavecol[8 * 128];         // per-wave column partials

    const int t    = threadIdx.x;
    const int lane = t & 31;
    const int w    = t >> 5;                   // wave id 0..7, 16 rows each
    const int rowbase = blockIdx.x * 128;

    // Stage 128x64 embeds tile into LDS, coalesced float4 loads.
    for (int i = t; i < 128 * 16; i += 256) {
        const int row = i >> 4;
        const int c4  = (i & 15) << 2;
        const float4 v = *(const float4*)(embeds + (size_t)(rowbase + row) * DIMS + c4);
        *(float4*)&lds_e[row * ROW_STRIDE + c4] = v;
    }
    __syncthreads();

    // Row norms (from LDS) and center norms (from global; centers stay L0/L2 resident).
    if (t < 128) {
        float se = 0.f;
        for (int d = 0; d < DIMS; ++d) { const float x = lds_e[t * ROW_STRIDE + d]; se += x * x; }
        e2[t] = se;
        float sc = 0.f;
        for (int d = 0; d < DIMS; ++d) { const float x = centers[t * DIMS + d]; sc += x * x; }
        c2[t] = sc;
    }
    __syncthreads();

    // WMMA fragment lane mapping (ISA 7.12.2, 32-bit A 16x4 / B 4x16):
    //   A: M = lane%16, VGPR0/1 = K pair; lanes 0-15 -> K{0,1}, lanes 16-31 -> K{2,3}
    //   B: N = lane%16, same K-pair split by half.
    const int m16  = lane & 15;
    const int hi   = lane >> 4;     // 0 lower half, 1 upper half
    const int dofs = hi * 2;

    float qv[8][8];                 // dist values: [column tile][C row reg]
    float rowsum[8];
    #pragma unroll
    for (int r = 0; r < 8; ++r) rowsum[r] = 0.f;

    #pragma unroll
    for (int ct = 0; ct < 8; ++ct) {
        const int k = ct * 16 + m16;          // center (column) index for this lane
        v8f acc = {0.f, 0.f, 0.f, 0.f, 0.f, 0.f, 0.f, 0.f};
        #pragma unroll
        for (int dc = 0; dc < 16; ++dc) {
            const int dbase = dc * 4 + dofs;  // even -> 8B aligned
            const float2 av = *(const float2*)&lds_e[(w * 16 + m16) * ROW_STRIDE + dbase];
            const float2 bv = *(const float2*)(centers + k * DIMS + dbase);
            v2f a; a.x = av.x; a.y = av.y;
            v2f b; b.x = bv.x; b.y = bv.y;
            acc = __builtin_amdgcn_wmma_f32_16x16x4_f32(
                /*neg_a=*/false, a, /*neg_b=*/false, b,
                /*c_mod=*/(short)0, acc, /*reuse_a=*/false, /*reuse_b=*/false);
        }
        // C/D layout: reg r, lane -> M = r + 8*hi, N = lane%16
        #pragma unroll
        for (int r = 0; r < 8; ++r) {
            const int ml = w * 16 + r + 8 * hi;           // local row in block
            float sq = e2[ml] + c2[k] - 2.0f * acc[r];
            sq = fmaxf(sq, 0.0f);
            const float dist = 1.0f / (1.0f + sq);        // student-t, alpha=1
            qv[ct][r] = dist;
            rowsum[r] += dist;
        }
    }

    // Row sums: reduce across the 16 lanes of each half (halves hold distinct rows).
    #pragma unroll
    for (int r = 0; r < 8; ++r) {
        float s = rowsum[r];
        s += __shfl_xor(s, 1, 32);
        s += __shfl_xor(s, 2, 32);
        s += __shfl_xor(s, 4, 32);
        s += __shfl_xor(s, 8, 32);
        rowsum[r] = s;
    }

    // Normalize, store q, build per-wave column partials.
    #pragma unroll
    for (int ct = 0; ct < 8; ++ct) {
        const int k = ct * 16 + m16;
        float colp = 0.f;
        #pragma unroll
        for (int r = 0; r < 8; ++r) {
            const float qq = qv[ct][r] / rowsum[r];
            const size_t grow = (size_t)(rowbase + w * 16 + r + 8 * hi);
            qout[grow * KC + k] = qq;
            colp += qq;
        }
        colp += __shfl_xor(colp, 16, 32);      // combine the two halves (16 rows total)
        if (!hi) wavecol[w * 128 + k] = colp;
    }
    __syncthreads();

    // Deterministic per-block column partial: fixed-order sum over the 8 waves.
    if (t < 128) {
        float s = 0.f;
        #pragma unroll
        for (int ww = 0; ww < 8; ++ww) s += wavecol[ww * 128 + t];
        colpart[(size_t)blockIdx.x * 128 + t] = s;
    }
}

// ---------------------------------------------------------------------------
// Colsum reduce: fixed-order sum over block partials (deterministic).
// ---------------------------------------------------------------------------
__global__ __launch_bounds__(128) void dec_colreduce(const float* __restrict__ colpart,
                                                     int nblk,
                                                     float* __restrict__ colsum)
{
    const int k = threadIdx.x;   // 128 threads
    float s = 0.f;
    for (int b = 0; b < nblk; ++b) s += colpart[(size_t)b * 128 + k];
    colsum[k] = s;
}

// ---------------------------------------------------------------------------
// Pass 2: one wave per row. w = q^2/colsum, p = w/S, loss terms p*(log w - log S - log q).
// ---------------------------------------------------------------------------
__global__ __launch_bounds__(256) void dec_pass2(const float* __restrict__ qin,    // d_out + 1
                                                 const float* __restrict__ colsum,
                                                 float* __restrict__ losspart)     // [gridDim.x]
{
    __shared__ float wloss[8];
    const int t = threadIdx.x, lane = t & 31, w = t >> 5;
    const size_t row = (size_t)blockIdx.x * 8 + w;
    const float* qr = qin + row * KC;

    float qs[4], ws[4];
    float S = 0.f;
    #pragma unroll
    for (int i = 0; i < 4; ++i) {
        const int k = lane + 32 * i;        // coalesced: 32 consecutive floats per load
        const float qq = qr[k];
        const float wv = qq * qq / colsum[k];
        qs[i] = qq; ws[i] = wv; S += wv;
    }
    S += __shfl_xor(S, 1, 32); S += __shfl_xor(S, 2, 32); S += __shfl_xor(S, 4, 32);
    S += __shfl_xor(S, 8, 32); S += __shfl_xor(S, 16, 32);
    const float lS = logf(S);

    float acc = 0.f;
    #pragma unroll
    for (int i = 0; i < 4; ++i) {
        const float p = ws[i] / S;
        acc += p * (logf(ws[i]) - lS - logf(qs[i]));   // p*(log p - log q)
    }
    acc += __shfl_xor(acc, 1, 32); acc += __shfl_xor(acc, 2, 32); acc += __shfl_xor(acc, 4, 32);
    acc += __shfl_xor(acc, 8, 32); acc += __shfl_xor(acc, 16, 32);

    if (lane == 0) wloss[w] = acc;
    __syncthreads();
    if (t == 0) {
        float s = 0.f;
        #pragma unroll
        for (int i = 0; i < 8; ++i) s += wloss[i];   // fixed order
        losspart[blockIdx.x] = s;
    }
}

// ---------------------------------------------------------------------------
// Finish: deterministic reduce of block loss partials; mean over N*K.
// ---------------------------------------------------------------------------
__global__ __launch_bounds__(256) void dec_finish(const float* __restrict__ losspart,
                                                  int n, float* __restrict__ out)
{
    __shared__ float red[256];
    const int t = threadIdx.x;
    float s = 0.f;
    for (int i = t; i < n; i += 256) s += losspart[i];   // fixed order per thread
    red[t] = s;
    __syncthreads();
    for (int st = 128; st > 0; st >>= 1) {
        if (t < st) red[t] += red[t + st];
        __syncthreads();
    }
    if (t == 0) out[0] = red[0] / (float)((size_t)N_ROWS * KC);
}

extern "C" void kernel_launch(void* const* d_in, const int* in_sizes, int n_in,
                              void* d_out, int out_size, void* d_ws, size_t ws_size,
                              hipStream_t stream) {
    const float* embeds  = (const float*)d_in[0];   // [400000, 64] fp32
    const float* centers = (const float*)d_in[1];   // [128, 64]    fp32
    float* out  = (float*)d_out;                    // [0]=loss, [1..]=q (N*K)
    float* qout = out + 1;
    float* ws   = (float*)d_ws;

    const int nblk1 = N_ROWS / 128;                 // 3125
    const int nblk2 = N_ROWS / 8;                   // 50000
    float* colpart  = ws;                           // 3125*128 floats
    float* colsum   = ws + (size_t)nblk1 * 128;     // 128 floats
    float* losspart = colsum + 128;                 // 50000 floats

    dec_pass1<<<nblk1, 256, 0, stream>>>(embeds, centers, qout, colpart);
    dec_colreduce<<<1, 128, 0, stream>>>(colpart, nblk1, colsum);
    dec_pass2<<<nblk2, 256, 0, stream>>>(qout, colsum, losspart);
    dec_finish<<<1, 256, 0, stream>>>(losspart, nblk2, out);
}